// RateAdaptionDecoder_36326833389762
// MI455X (gfx1250) — compile-verified
//
#include <hip/hip_runtime.h>

// Problem constants (match reference)
#define BB 16
#define CC 256
#define SS 256
#define RN 6
#define NTOK 4096          // B*S tokens
#define TILE_M 16
#define MAX_TILES 256      // NTOK / TILE_M

typedef float v2f __attribute__((ext_vector_type(2)));
typedef float v8f __attribute__((ext_vector_type(8)));

// ---------------- phase 1: group tokens by rate index ----------------

__global__ void ra_zero_cnt(int* __restrict__ cnt) {
    if (threadIdx.x < RN) cnt[threadIdx.x] = 0;
}

__global__ void ra_scatter(const int* __restrict__ idx,
                           int* __restrict__ cnt,
                           int* __restrict__ lists) {
    int t = blockIdx.x * blockDim.x + threadIdx.x;
    if (t < NTOK) {
        int r = idx[t];
        int pos = atomicAdd(&cnt[r], 1);
        lists[r * NTOK + pos] = t;   // token id = b*S + s
    }
}

// ---------------- phase 2: grouped GEMM with f32 WMMA ----------------
// grid.x = rate * MAX_TILES + tokenTile; block = 256 threads = 8 waves.
// Each block computes a full 16 x 256 output tile: wave w owns columns
// [w*16, w*16+16) and [128 + w*16, 128 + w*16 + 16)  (two WMMAs per K-step,
// sharing one A fragment -> x is staged/read exactly once per token).

__global__ __launch_bounds__(256)
void ra_gemm(const float* __restrict__ x,      // (B, C, S)
             const float* __restrict__ w,      // (RN, C, C) row-major (k, c)
             const float* __restrict__ bias,   // (RN, C)
             const int*   __restrict__ cnt,
             const int*   __restrict__ lists,
             float* __restrict__ out) {        // (B, C, S)
    __shared__ float xs[TILE_M][CC + 1];       // +1 pad: bank-conflict free
    __shared__ int   stok[TILE_M];

    const int r     = blockIdx.x >> 8;         // / MAX_TILES
    const int tile  = blockIdx.x & (MAX_TILES - 1);
    const int count = cnt[r];
    const int base  = tile * TILE_M;
    if (base >= count) return;                 // whole block exits: EXEC stays uniform

    const int tid  = threadIdx.x;
    const int wave = tid >> 5;
    const int lane = tid & 31;
    const int l16  = lane & 15;
    const int hi   = lane >> 4;                // 0: K lanes {0,1}, 1: K lanes {2,3}

    if (tid < TILE_M) {
        int p = base + tid;
        stok[tid] = (p < count) ? lists[r * NTOK + p] : -1;
    }
    __syncthreads();

    // Stage 16 token vectors (length K=256) into LDS; zero-fill invalid rows.
    {
        int row = tid >> 4;                    // 0..15
        int kc  = tid & 15;                    // 16-wide k chunk id
        int tok = stok[row];
        if (tok >= 0) {
            int b = tok >> 8, s = tok & 255;
            const float* xp = x + (size_t)b * (CC * SS) + s;
            #pragma unroll
            for (int j = 0; j < 16; ++j) {
                int k = kc * 16 + j;
                xs[row][k] = xp[(size_t)k * SS];
            }
        } else {
            #pragma unroll
            for (int j = 0; j < 16; ++j) xs[row][kc * 16 + j] = 0.0f;
        }
    }
    __syncthreads();

    // This lane's two output columns (= weight columns = bias/residual indices)
    const int n0 = wave * 16 + l16;
    const int n1 = n0 + 128;

    // Fold bias into the accumulators: bias depends only on the column, which
    // is identical across all 8 C VGPRs of a lane.
    const float bv0 = bias[r * CC + n0];
    const float bv1 = bias[r * CC + n1];
    v8f acc0, acc1;
    #pragma unroll
    for (int i = 0; i < 8; ++i) { acc0[i] = bv0; acc1[i] = bv1; }

    const float* wp0 = w + (size_t)r * (CC * CC) + n0;   // row k stride = CC
    const float* wp1 = w + (size_t)r * (CC * CC) + n1;

    // K = 256 -> 64 steps, 2x V_WMMA_F32_16X16X4_F32 per step (shared A)
    #pragma unroll 8
    for (int kk = 0; kk < CC / 4; ++kk) {
        const int kbase = kk * 4 + hi * 2;
        v2f a, b0, b1;
        // A 16x4 layout: lanes 0-15 row M=l16 K={0,1}; lanes 16-31 K={2,3}
        a.x = xs[l16][kbase + 0];
        a.y = xs[l16][kbase + 1];
        // B 4x16 layout (mirrored): lanes 0-15 col N, K={0,1}; lanes 16-31 K={2,3}
        b0.x = wp0[(size_t)(kbase + 0) * CC];
        b0.y = wp0[(size_t)(kbase + 1) * CC];
        b1.x = wp1[(size_t)(kbase + 0) * CC];
        b1.y = wp1[(size_t)(kbase + 1) * CC];
        acc0 = __builtin_amdgcn_wmma_f32_16x16x4_f32(
            false, a, false, b0, (short)0, acc0, false, false);
        acc1 = __builtin_amdgcn_wmma_f32_16x16x4_f32(
            false, a, false, b1, (short)0, acc1, false, false);
    }

    // Epilogue: C/D 16x16 f32 layout -> VGPR v: row m = v + hi*8, col = l16.
    // y[b, n, s] = x_token[n] (residual, from LDS) + acc (= dot + bias)
    #pragma unroll
    for (int v = 0; v < 8; ++v) {
        int m   = v + hi * 8;
        int tok = stok[m];
        if (tok >= 0) {
            int b = tok >> 8, s = tok & 255;
            size_t obase = (size_t)b * (CC * SS) + s;
            out[obase + (size_t)n0 * SS] = acc0[v] + xs[m][n0];
            out[obase + (size_t)n1 * SS] = acc1[v] + xs[m][n1];
        }
    }
}

// ---------------- launcher ----------------

extern "C" void kernel_launch(void* const* d_in, const int* in_sizes, int n_in,
                              void* d_out, int out_size, void* d_ws, size_t ws_size,
                              hipStream_t stream) {
    const float* x    = (const float*)d_in[0];   // (16, 256, 256) f32
    const float* wgt  = (const float*)d_in[1];   // (6, 256, 256) f32
    const float* bias = (const float*)d_in[2];   // (6, 256) f32
    const int*   idx  = (const int*)d_in[3];     // (4096,) i32
    float* out = (float*)d_out;

    int* cnt   = (int*)d_ws;        // 6 cursors (16-int slot for alignment)
    int* lists = cnt + 16;          // 6 * 4096 token ids (~98 KB total ws use)

    ra_zero_cnt<<<1, 64, 0, stream>>>(cnt);
    ra_scatter<<<NTOK / 256, 256, 0, stream>>>(idx, cnt, lists);

    ra_gemm<<<RN * MAX_TILES, 256, 0, stream>>>(x, wgt, bias, cnt, lists, out);
}